// FAquantizer_58145267254039
// MI455X (gfx1250) — compile-verified
//
#include <hip/hip_runtime.h>
#include <cstddef>
#include <cstdint>

typedef float v2f __attribute__((ext_vector_type(2)));
typedef float v4f __attribute__((ext_vector_type(4)));
typedef float v8f __attribute__((ext_vector_type(8)));

#define B_ 8
#define C_ 1024
#define T_ 4096
#define K_ 1024
#define D_ 8
#define TILE 16
#define NTH 128

// ---- LDS layout (float offsets) ----
#define OFF_RES  0          // [1024][16] current residual
#define OFF_OUTS 16384      // [1024][16] outs accumulator
#define OFF_ZACC 32768      // [1024][16] per-RVQ z accumulator
#define OFF_IW   49152      // [8][1024]  staged in_proj weight
#define OFF_CB   57344      // [1024][8]  staged codebook
#define OFF_OW   65536      // [1024][8]  staged out_proj weight
#define OFF_OB   73728      // [1024]     staged out_proj bias
#define OFF_IB   74752      // [8]        staged in_proj bias
#define OFF_ZE   74760      // [8..16][16] z_e (d-major)
#define OFF_ZQ   75016      // [8..16][16] z_q (d-major)
#define OFF_CBRN 75272      // [1024] codebook inverse norms
#define OFF_PART 76296      // [4][256] in_proj partial tiles
#define OFF_REDV 77320      // [128] argmax values
#define OFF_REDI 77448      // [128] argmax indices (int)
#define OFF_CM   77576      // [1] commit accumulator
#define SMEM_FLOATS 77580
#define SMEM_BYTES  (SMEM_FLOATS * 4)

static __device__ __forceinline__ v8f wmma_f32(v2f a, v2f b, v8f c) {
  // D(16x16,f32) = A(16x4,f32) * B(4x16,f32) + C
  return __builtin_amdgcn_wmma_f32_16x16x4_f32(false, a, false, b, (short)0, c,
                                               false, false);
}

// ---- CDNA5 async global->LDS copy (ASYNCcnt-tracked DMA into LDS) ----
static __device__ __forceinline__ void async_copy_b128(uint32_t lds_byte_addr,
                                                       const float* gaddr) {
  asm volatile("global_load_async_to_lds_b128 %0, %1, off"
               :
               : "v"(lds_byte_addr), "v"(gaddr)
               : "memory");
}
static __device__ __forceinline__ void wait_async0() {
  asm volatile("s_wait_asynccnt 0x0" ::: "memory");
}
static __device__ __forceinline__ uint32_t lds_addr_of(const float* p) {
  // low 32 bits of a flat pointer into the LDS aperture == LDS byte offset
  return (uint32_t)(uintptr_t)p;
}

static __device__ __forceinline__ void stage_async(float* dst_lds,
                                                   const float* src,
                                                   int nfloats, int tid) {
  uint32_t base = lds_addr_of(dst_lds);
  for (int e = tid; e < (nfloats >> 2); e += NTH)
    async_copy_b128(base + e * 16, src + e * 4);
}

static __device__ __forceinline__ void prefetch_ic(float* smem,
                                                   const float* iw,
                                                   const float* cb,
                                                   const float* ibb, int tid) {
  stage_async(smem + OFF_IW, iw, D_ * C_, tid);
  stage_async(smem + OFF_CB, cb, K_ * D_, tid);
  if (tid < 2)
    async_copy_b128(lds_addr_of(smem + OFF_IB) + tid * 16, ibb + tid * 4);
}
static __device__ __forceinline__ void prefetch_oo(float* smem,
                                                   const float* ow,
                                                   const float* ob, int tid) {
  stage_async(smem + OFF_OW, ow, C_ * D_, tid);
  stage_async(smem + OFF_OB, ob, C_, tid);
}

// One quantizer stage; all current-stage weights already staged in LDS.
// Issues async prefetch for the next stage's weights once buffers go dead.
static __device__ __forceinline__ void do_stage(
    float* smem, int tid, const float* niw, const float* ncb, const float* nib,
    const float* nowp, const float* nob, bool hasNext) {
  float* RESs  = smem + OFF_RES;
  float* ZACCs = smem + OFF_ZACC;
  float* IWs   = smem + OFF_IW;
  float* CBs   = smem + OFF_CB;
  float* OWs   = smem + OFF_OW;
  float* OBs   = smem + OFF_OB;
  float* IBs   = smem + OFF_IB;
  float* ZEs   = smem + OFF_ZE;
  float* ZQs   = smem + OFF_ZQ;
  float* CBRNs = smem + OFF_CBRN;
  float* PARTs = smem + OFF_PART;
  float* REDVs = smem + OFF_REDV;
  int*   REDIs = (int*)(smem + OFF_REDI);
  float* CMs   = smem + OFF_CM;

  const int wv = tid >> 5, lane = tid & 31, tL = lane & 15, hi = lane >> 4;

  // weights for this stage were async-prefetched; complete them
  wait_async0();
  __syncthreads();

  // ---- codebook inverse norms (from staged LDS codebook)
  for (int j = 0; j < K_ / NTH; ++j) {
    int code = tid + j * NTH;
    v4f c0 = *(const v4f*)(CBs + code * D_);
    v4f c1 = *(const v4f*)(CBs + code * D_ + 4);
    float s = c0.x * c0.x + c0.y * c0.y + c0.z * c0.z + c0.w * c0.w +
              c1.x * c1.x + c1.y * c1.y + c1.z * c1.z + c1.w * c1.w;
    CBRNs[code] = 1.0f / fmaxf(sqrtf(s), 1e-12f);
  }
  __syncthreads();

  // ---- in_proj: z_e^T(16t x 16d) = res^T(16t x C) * in_w^T(C x 16d)
  // K = C = 1024 split across 4 waves; dual accumulators pipeline the WMMAs.
  // Lanes with d>=8 read row (d&7): duplicate columns the reduce ignores.
  {
    v8f acc0 = {0.f, 0.f, 0.f, 0.f, 0.f, 0.f, 0.f, 0.f};
    v8f acc1 = {0.f, 0.f, 0.f, 0.f, 0.f, 0.f, 0.f, 0.f};
    const float* iwrow = IWs + (size_t)(tL & 7) * C_;
    for (int kk = 0; kk < 64; kk += 2) {
      int cb0 = (wv * 64 + kk) * 4 + 2 * hi;
      int cb1 = cb0 + 4;
      v2f a0, a1;
      a0.x = RESs[(cb0 + 0) * TILE + tL];
      a0.y = RESs[(cb0 + 1) * TILE + tL];
      a1.x = RESs[(cb1 + 0) * TILE + tL];
      a1.y = RESs[(cb1 + 1) * TILE + tL];
      v2f b0 = *(const v2f*)(iwrow + cb0);
      v2f b1 = *(const v2f*)(iwrow + cb1);
      acc0 = wmma_f32(a0, b0, acc0);
      acc1 = wmma_f32(a1, b1, acc1);
    }
#pragma unroll
    for (int r = 0; r < 8; ++r)
      PARTs[wv * 256 + r * 32 + lane] = acc0[r] + acc1[r];
  }
  __syncthreads();
  for (int p = tid; p < 256; p += NTH) {
    float s = PARTs[p] + PARTs[p + 256] + PARTs[p + 512] + PARTs[p + 768];
    int ln = p & 31, r = p >> 5;
    int tt = r + 8 * (ln >> 4);
    int dd = ln & 15;
    if (dd < 8) ZEs[dd * TILE + tt] = s + IBs[dd];
  }
  __syncthreads();

  // ---- scores + argmax: scores(16k x 16t) = cb_n(16k x 8d) * z_e(8d x 16t)
  {
    float bestv = -3.4e38f;
    int besti = 0;
#pragma unroll 2
    for (int j = 0; j < 16; ++j) {
      int kbase = (wv * 16 + j) * 16;
      int code = kbase + tL;
      float rn = CBRNs[code];
      v8f acc = {0.f, 0.f, 0.f, 0.f, 0.f, 0.f, 0.f, 0.f};
#pragma unroll
      for (int s = 0; s < 2; ++s) {
        int dbase = 4 * s + 2 * hi;
        v2f a = *(const v2f*)(CBs + code * D_ + dbase);
        a.x *= rn;
        a.y *= rn;
        v2f bb;
        bb.x = ZEs[(dbase + 0) * TILE + tL];
        bb.y = ZEs[(dbase + 1) * TILE + tL];
        acc = wmma_f32(a, bb, acc);
      }
#pragma unroll
      for (int r = 0; r < 8; ++r) {
        int idx = kbase + r + 8 * hi;
        float v = acc[r];
        if (v > bestv || (v == bestv && idx < besti)) {
          bestv = v;
          besti = idx;
        }
      }
    }
    REDVs[tid] = bestv;
    REDIs[tid] = besti;
  }
  __syncthreads();
  if (tid < 16) {  // per-column final argmax, z_q gather, commit accumulation
    float bestv = -3.4e38f;
    int besti = 0;
    for (int w4 = 0; w4 < 4; ++w4) {
#pragma unroll
      for (int h = 0; h < 32; h += 16) {
        int p = w4 * 32 + tid + h;
        float v = REDVs[p];
        int idx = REDIs[p];
        if (v > bestv || (v == bestv && idx < besti)) {
          bestv = v;
          besti = idx;
        }
      }
    }
    v4f c0 = *(const v4f*)(CBs + (size_t)besti * D_);
    v4f c1 = *(const v4f*)(CBs + (size_t)besti * D_ + 4);
    float cq[8] = {c0.x, c0.y, c0.z, c0.w, c1.x, c1.y, c1.z, c1.w};
    float csum = 0.f;
#pragma unroll
    for (int d = 0; d < 8; ++d) {
      float ze = ZEs[d * TILE + tid];
      float df = ze - cq[d];
      csum += df * df;  // forward commit == cbl
      ZQs[d * TILE + tid] = cq[d];
    }
    atomicAdd(CMs, csum);
  }
  __syncthreads();

  // IW/CB/IB buffers are dead now: overlap next stage's copy with out_proj.
  if (hasNext) prefetch_ic(smem, niw, ncb, nib, tid);

  // ---- out_proj: zq_out(16c x 16t) = out_w(16c x 8d) * z_q(8d x 16t)
  {
#pragma unroll 2
    for (int j = 0; j < 16; ++j) {
      int cbase = (wv * 16 + j) * 16;
      v8f acc = {0.f, 0.f, 0.f, 0.f, 0.f, 0.f, 0.f, 0.f};
      const float* owp = OWs + (size_t)(cbase + tL) * D_;
#pragma unroll
      for (int s = 0; s < 2; ++s) {
        int dbase = 4 * s + 2 * hi;
        v2f a = *(const v2f*)(owp + dbase);
        v2f bb;
        bb.x = ZQs[(dbase + 0) * TILE + tL];
        bb.y = ZQs[(dbase + 1) * TILE + tL];
        acc = wmma_f32(a, bb, acc);
      }
      v4f ob0 = *(const v4f*)(OBs + cbase + 8 * hi);
      v4f ob1 = *(const v4f*)(OBs + cbase + 8 * hi + 4);
      float obv[8] = {ob0.x, ob0.y, ob0.z, ob0.w, ob1.x, ob1.y, ob1.z, ob1.w};
#pragma unroll
      for (int r = 0; r < 8; ++r) {
        int c = cbase + r + 8 * hi;
        float v = acc[r] + obv[r];
        int addr = c * TILE + tL;
        RESs[addr] -= v;
        ZACCs[addr] += v;
      }
    }
  }
  __syncthreads();

  // OW/OB dead: prefetch next stage's out_proj weights during boundary/stage.
  if (hasNext) prefetch_oo(smem, nowp, nob, tid);
}

__global__ void __launch_bounds__(NTH) rvq_fused_kernel(
    const float* __restrict__ x, const int* __restrict__ noise_added,
    const int* __restrict__ recon_noisy, const float* __restrict__ res_rand,
    const float* p_iw, const float* p_ib, const float* p_cb, const float* p_ow, const float* p_ob,
    const float* c_iw, const float* c_ib, const float* c_cb, const float* c_ow, const float* c_ob,
    const float* t_iw, const float* t_ib, const float* t_cb, const float* t_ow, const float* t_ob,
    const float* r_iw, const float* r_ib, const float* r_cb, const float* r_ow, const float* r_ob,
    float* __restrict__ out) {
  extern __shared__ float smem[];
  const int tid = threadIdx.x;
  const int wg = blockIdx.x;
  const int b = wg >> 8;            // T/TILE = 256 tiles per batch
  const int t0 = (wg & 255) * TILE;

  const size_t NPER = (size_t)B_ * C_ * T_;  // 33554432
  const size_t base = ((size_t)b * C_) * T_ + t0;
  float* outs_g = out + base;
  float* zp_g = out + 1 * NPER + base;
  float* zc_g = out + 2 * NPER + base;
  float* zt_g = out + 3 * NPER + base;
  float* zr_g = out + 4 * NPER + base;
  float* cm_g = out + 5 * NPER;  // commit, then cbl at +1

  // residual mask for this batch
  const int na = noise_added[b] != 0;
  const int rn = recon_noisy[b] != 0;
  const float mask = na ? (rn ? 1.0f : 0.0f) : res_rand[b];

  float* RESs  = smem + OFF_RES;
  float* OUTSs = smem + OFF_OUTS;
  float* ZACCs = smem + OFF_ZACC;

  const int cg = tid >> 2;
  const int t4 = (tid & 3) * 4;
  const v4f vzero = {0.f, 0.f, 0.f, 0.f};

  // x tile lives in registers (32 v4f per thread); frees LDS for weights.
  v4f Xr[32];

  // kick off stage-0 weight staging before touching x
  prefetch_ic(smem, p_iw, p_cb, p_ib, tid);
  prefetch_oo(smem, p_ow, p_ob, tid);

  // ---- init: load x tile -> regs, RES=X, OUTS=0, ZACC=0
#pragma unroll
  for (int j = 0; j < 32; ++j) {
    int c = cg + j * 32;
    v4f xv = *(const v4f*)(x + ((size_t)b * C_ + c) * T_ + t0 + t4);
    int a = c * TILE + t4;
    Xr[j] = xv;
    *(v4f*)(RESs + a) = xv;
    *(v4f*)(OUTSs + a) = vzero;
    *(v4f*)(ZACCs + a) = vzero;
  }
  if (tid == 0) smem[OFF_CM] = 0.f;
  __syncthreads();

  // RVQ boundary pass (elementwise over the tiles; X lives in Xr registers)
  // mode 0 (after p): z->global, OUTS+=Z, RES=X, Z=0
  // mode 1 (after c): z->global, OUTS+=Z, X=X-OUTS (tim), RES=X, Z=0
  // mode 2 (after t): z->global, OUTS+=Z, X=X-Z (resid), RES=X, Z=0
  // mode 3 (after r): z->global, OUTS+=mask*Z, OUTS->global
  auto bpass = [&](float* zglob, int mode, float mk, float* outsglob) {
#pragma unroll
    for (int j = 0; j < 32; ++j) {
      int c = cg + j * 32;
      int a = c * TILE + t4;
      v4f zv = *(v4f*)(ZACCs + a);
      *(v4f*)(zglob + (size_t)c * T_ + t4) = zv;
      v4f ov = *(v4f*)(OUTSs + a);
      if (mode == 3)
        ov += zv * mk;
      else
        ov += zv;
      *(v4f*)(OUTSs + a) = ov;
      if (mode == 3) {
        *(v4f*)(outsglob + (size_t)c * T_ + t4) = ov;
      } else {
        v4f xv = Xr[j];
        if (mode == 1) xv -= ov;  // tim = x - (z_p + z_c)
        if (mode == 2) xv -= zv;  // resid = tim - z_t
        Xr[j] = xv;
        *(v4f*)(RESs + a) = xv;
        *(v4f*)(ZACCs + a) = vzero;
      }
    }
  };

  // ---- RVQ p (1 stage)
  do_stage(smem, tid, c_iw, c_cb, c_ib, c_ow, c_ob, true);
  bpass(zp_g, 0, 0.f, nullptr);
  __syncthreads();

  // ---- RVQ c (2 stages, restarts from x)
  do_stage(smem, tid, c_iw + D_ * C_, c_cb + K_ * D_, c_ib + D_,
           c_ow + C_ * D_, c_ob + C_, true);
  do_stage(smem, tid, t_iw, t_cb, t_ib, t_ow, t_ob, true);
  bpass(zc_g, 1, 0.f, nullptr);
  __syncthreads();

  // ---- RVQ t (2 stages on tim)
  do_stage(smem, tid, t_iw + D_ * C_, t_cb + K_ * D_, t_ib + D_,
           t_ow + C_ * D_, t_ob + C_, true);
  do_stage(smem, tid, r_iw, r_cb, r_ib, r_ow, r_ob, true);
  bpass(zt_g, 2, 0.f, nullptr);
  __syncthreads();

  // ---- RVQ r (3 stages on resid)
  do_stage(smem, tid, r_iw + D_ * C_, r_cb + K_ * D_, r_ib + D_,
           r_ow + C_ * D_, r_ob + C_, true);
  do_stage(smem, tid, r_iw + 2 * D_ * C_, r_cb + 2 * K_ * D_, r_ib + 2 * D_,
           r_ow + 2 * C_ * D_, r_ob + 2 * C_, true);
  do_stage(smem, tid, nullptr, nullptr, nullptr, nullptr, nullptr, false);
  bpass(zr_g, 3, mask, outs_g);
  __syncthreads();

  // ---- commit / cbl (identical in forward): one atomic per WG
  if (tid == 0) {
    float cm = smem[OFF_CM] * (1.0f / ((float)B_ * (float)D_ * (float)T_));
    atomicAdd(cm_g + 0, cm);
    atomicAdd(cm_g + 1, cm);
  }
}

__global__ void zero_scalars_kernel(float* __restrict__ p) {
  if (threadIdx.x < 2) p[threadIdx.x] = 0.f;
}

extern "C" void kernel_launch(void* const* d_in, const int* in_sizes, int n_in,
                              void* d_out, int out_size, void* d_ws,
                              size_t ws_size, hipStream_t stream) {
  (void)in_sizes; (void)n_in; (void)out_size; (void)d_ws; (void)ws_size;
  const float* x = (const float*)d_in[0];
  // d_in[1] = wave_segments (unused in forward)
  const int* noise_added = (const int*)d_in[2];
  const int* recon_noisy = (const int*)d_in[3];
  const float* res_rand = (const float*)d_in[4];
  const float* p_iw = (const float*)d_in[5];
  const float* p_ib = (const float*)d_in[6];
  const float* p_cb = (const float*)d_in[7];
  const float* p_ow = (const float*)d_in[8];
  const float* p_ob = (const float*)d_in[9];
  const float* c_iw = (const float*)d_in[10];
  const float* c_ib = (const float*)d_in[11];
  const float* c_cb = (const float*)d_in[12];
  const float* c_ow = (const float*)d_in[13];
  const float* c_ob = (const float*)d_in[14];
  const float* t_iw = (const float*)d_in[15];
  const float* t_ib = (const float*)d_in[16];
  const float* t_cb = (const float*)d_in[17];
  const float* t_ow = (const float*)d_in[18];
  const float* t_ob = (const float*)d_in[19];
  const float* r_iw = (const float*)d_in[20];
  const float* r_ib = (const float*)d_in[21];
  const float* r_cb = (const float*)d_in[22];
  const float* r_ow = (const float*)d_in[23];
  const float* r_ob = (const float*)d_in[24];
  float* out = (float*)d_out;

  const size_t NPER = (size_t)B_ * C_ * T_;
  zero_scalars_kernel<<<1, 32, 0, stream>>>(out + 5 * NPER);

  dim3 grid((B_ * T_) / TILE);  // 2048 workgroups
  dim3 block(NTH);              // 4 wave32
  rvq_fused_kernel<<<grid, block, SMEM_BYTES, stream>>>(
      x, noise_added, recon_noisy, res_rand, p_iw, p_ib, p_cb, p_ow, p_ob, c_iw,
      c_ib, c_cb, c_ow, c_ob, t_iw, t_ib, t_cb, t_ow, t_ob, r_iw, r_ib, r_cb,
      r_ow, r_ob, out);
}